// GAT1_34230889349736
// MI455X (gfx1250) — compile-verified
//
#include <hip/hip_runtime.h>
#include <math.h>

// ---------------------------------------------------------------------------
// GNN (GAT + SAGPool x3) forward for MI455X / gfx1250, wave32.
// Dense GEMMs use V_WMMA_F32_16X16X4_F32 (full f32 precision; the pipeline is
// bandwidth-bound at 23.3 TB/s so f32 WMMA throughput is not the limiter).
// Edge softmax/scatter uses wave-per-edge atomics (64-wide messages).
// BatchNorm uses a deterministic two-stage reduction.
// Top-k uses a per-graph bitonic sort in LDS.
// ---------------------------------------------------------------------------

typedef __attribute__((ext_vector_type(2))) float v2f;
typedef __attribute__((ext_vector_type(8))) float v8f;

#define HIDDIM 64
#define NGRAPH 32

__device__ __forceinline__ float lrelu02(float x) { return x >= 0.0f ? x : 0.2f * x; }

// monotone float -> uint map so unsigned atomicMax == float max
__device__ __forceinline__ unsigned fenc(float f) {
  unsigned b = __float_as_uint(f);
  return (b & 0x80000000u) ? ~b : (b | 0x80000000u);
}
__device__ __forceinline__ float fdec(unsigned u) {
  unsigned b = (u & 0x80000000u) ? (u ^ 0x80000000u) : ~u;
  return __uint_as_float(b);
}

// ---------------------------------------------------------------------------
// C[M x 64] = A[M x K] @ B[K x 64] (+ bias), K % 4 == 0, M % 128 == 0.
// One wave computes a 16-row x 64-col tile via 4 f32 WMMA accumulators.
// f32 16x16x4 WMMA layouts (ISA 7.12.2):
//   A 16x4 : lane l holds row M=l&15, K = {2*(l>>4), 2*(l>>4)+1} in 2 VGPRs
//   B 4x16 : lane l holds col N=l&15, K = {2*(l>>4), 2*(l>>4)+1} in 2 VGPRs
//   C 16x16: VGPR v, lane l -> M = (l>>4)*8 + v, N = l&15
// ---------------------------------------------------------------------------
__global__ __launch_bounds__(256) void gnn_gemm_n64(
    const float* __restrict__ A, const float* __restrict__ B,
    const float* __restrict__ bias, float* __restrict__ C,
    int M, int K)
{
  const int lane = threadIdx.x & 31;
  const int wave = threadIdx.x >> 5;
  const int row0 = blockIdx.x * 128 + wave * 16;
  if (row0 >= M) return;
  const int half = lane >> 4;   // 0 or 1
  const int l15  = lane & 15;

  v8f acc0 = {}, acc1 = {}, acc2 = {}, acc3 = {};
  const float* Arow = A + (size_t)(row0 + l15) * K;

  for (int k0 = 0; k0 < K; k0 += 4) {
    if ((k0 & 63) == 0) __builtin_prefetch(Arow + k0 + 64, 0, 1);  // global_prefetch_b8
    const int ka = k0 + 2 * half;
    v2f a = *(const v2f*)(Arow + ka);          // 8B-aligned (ka even)
    const float* B0 = B + (size_t)ka * HIDDIM; // row ka
    const float* B1 = B0 + HIDDIM;             // row ka+1
    v2f b0, b1, b2, b3;
    b0.x = B0[l15];      b0.y = B1[l15];
    b1.x = B0[16 + l15]; b1.y = B1[16 + l15];
    b2.x = B0[32 + l15]; b2.y = B1[32 + l15];
    b3.x = B0[48 + l15]; b3.y = B1[48 + l15];
    acc0 = __builtin_amdgcn_wmma_f32_16x16x4_f32(false, a, false, b0, (short)0, acc0, false, false);
    acc1 = __builtin_amdgcn_wmma_f32_16x16x4_f32(false, a, false, b1, (short)0, acc1, false, false);
    acc2 = __builtin_amdgcn_wmma_f32_16x16x4_f32(false, a, false, b2, (short)0, acc2, false, false);
    acc3 = __builtin_amdgcn_wmma_f32_16x16x4_f32(false, a, false, b3, (short)0, acc3, false, false);
  }

  float bv0 = 0.f, bv1 = 0.f, bv2 = 0.f, bv3 = 0.f;
  if (bias) { bv0 = bias[l15]; bv1 = bias[16 + l15]; bv2 = bias[32 + l15]; bv3 = bias[48 + l15]; }

  float* Crow = C + (size_t)(row0 + half * 8) * HIDDIM;
#pragma unroll
  for (int v = 0; v < 8; ++v) {
    float* cr = Crow + (size_t)v * HIDDIM;
    cr[l15]      = acc0[v] + bv0;
    cr[16 + l15] = acc1[v] + bv1;
    cr[32 + l15] = acc2[v] + bv2;
    cr[48 + l15] = acc3[v] + bv3;
  }
}

// ---- fills --------------------------------------------------------------
__global__ void gnn_fill_f32(float* p, float v, long long n) {
  long long i = (long long)blockIdx.x * blockDim.x + threadIdx.x;
  if (i < n) p[i] = v;
}
__global__ void gnn_fill_u32(unsigned* p, unsigned v, long long n) {
  long long i = (long long)blockIdx.x * blockDim.x + threadIdx.x;
  if (i < n) p[i] = v;
}
__global__ void gnn_fill_i32(int* p, int v, long long n) {
  long long i = (long long)blockIdx.x * blockDim.x + threadIdx.x;
  if (i < n) p[i] = v;
}

// ---- per-node dual dot (es/ed): wave per node ---------------------------
__global__ __launch_bounds__(256) void gnn_dual_dot(
    const float* __restrict__ X, const float* __restrict__ va, const float* __restrict__ vb,
    float* __restrict__ outa, float* __restrict__ outb, int N)
{
  int lane = threadIdx.x & 31;
  int node = (int)((blockIdx.x * (long long)blockDim.x + threadIdx.x) >> 5);
  if (node >= N) return;
  const float* xr = X + (size_t)node * HIDDIM;
  float x0 = xr[lane], x1 = xr[lane + 32];
  float sa = x0 * va[lane] + x1 * va[lane + 32];
  float sb = x0 * vb[lane] + x1 * vb[lane + 32];
#pragma unroll
  for (int off = 16; off > 0; off >>= 1) {
    sa += __shfl_xor(sa, off, 32);
    sb += __shfl_xor(sb, off, 32);
  }
  if (lane == 0) { outa[node] = sa; outb[node] = sb; }
}

// ---- GAT edge pass 1: segment max of leaky_relu(es[s]+ed[d]) ------------
__global__ __launch_bounds__(256) void gnn_edge_max(
    const int* __restrict__ src, const int* __restrict__ dst,
    const unsigned char* __restrict__ mask,
    const float* __restrict__ es, const float* __restrict__ ed,
    unsigned* __restrict__ emax, int E, int N)
{
  int i = blockIdx.x * blockDim.x + threadIdx.x;
  if (i >= E + N) return;
  int s, d;
  if (i < E) {
    if (mask && !mask[i]) return;
    s = src[i]; d = dst[i];
  } else { s = d = i - E; }   // self-loop
  atomicMax(&emax[d], fenc(lrelu02(es[s] + ed[d])));
}

// ---- GAT edge pass 2: unnormalized softmax scatter (wave per edge) ------
__global__ __launch_bounds__(256) void gnn_edge_accum(
    const int* __restrict__ src, const int* __restrict__ dst,
    const unsigned char* __restrict__ mask,
    const float* __restrict__ es, const float* __restrict__ ed,
    const unsigned* __restrict__ emax, const float* __restrict__ XL,
    float* __restrict__ denom, float* __restrict__ XG, int E, int N)
{
  int lane = threadIdx.x & 31;
  int i = (int)((blockIdx.x * (long long)blockDim.x + threadIdx.x) >> 5);
  if (i >= E + N) return;
  int s, d;
  if (i < E) {
    if (mask && !mask[i]) return;
    s = src[i]; d = dst[i];
  } else { s = d = i - E; }
  float e = lrelu02(es[s] + ed[d]);
  float p = expf(e - fdec(emax[d]));
  if (lane == 0) atomicAdd(&denom[d], p);
  const float* xs = XL + (size_t)s * HIDDIM;
  float* xd = XG + (size_t)d * HIDDIM;
  atomicAdd(&xd[lane],      p * xs[lane]);
  atomicAdd(&xd[lane + 32], p * xs[lane + 32]);
}

// ---- GAT finalize: XG = XG/denom + bias ---------------------------------
__global__ void gnn_gat_finalize(float* XG, const float* __restrict__ denom,
                                 const float* __restrict__ bg, long long NE)
{
  long long i = (long long)blockIdx.x * blockDim.x + threadIdx.x;
  if (i >= NE) return;
  int n = (int)(i >> 6), c = (int)(i & 63);
  XG[i] = XG[i] / denom[n] + bg[c];
}

// ---- BatchNorm: deterministic two-stage reduction -----------------------
__global__ __launch_bounds__(256) void gnn_bn_partial(
    const float* __restrict__ X, float* __restrict__ pS, float* __restrict__ pQ, int N)
{
  __shared__ float sS[256], sQ[256];
  int c  = threadIdx.x & 63;
  int rg = threadIdx.x >> 6;          // 0..3
  int row0 = blockIdx.x * 256;
  float s = 0.f, q = 0.f;
  for (int r = rg; r < 256; r += 4) {
    int row = row0 + r;
    if (row < N) {
      float v = X[(size_t)row * HIDDIM + c];
      s += v; q += v * v;
    }
  }
  sS[threadIdx.x] = s; sQ[threadIdx.x] = q;
  __syncthreads();
  if (rg == 0) {
    s = sS[c] + sS[c + 64] + sS[c + 128] + sS[c + 192];
    q = sQ[c] + sQ[c + 64] + sQ[c + 128] + sQ[c + 192];
    pS[blockIdx.x * 64 + c] = s;
    pQ[blockIdx.x * 64 + c] = q;
  }
}

__global__ void gnn_bn_finalize(const float* __restrict__ pS, const float* __restrict__ pQ,
                                int nb, int N, const float* __restrict__ g,
                                float* __restrict__ mean, float* __restrict__ scale)
{
  int c = threadIdx.x;   // 64 threads
  float s = 0.f, q = 0.f;
  for (int b = 0; b < nb; ++b) { s += pS[b * 64 + c]; q += pQ[b * 64 + c]; }
  float m = s / (float)N;
  float v = q / (float)N - m * m;
  mean[c]  = m;
  scale[c] = g[c] / sqrtf(v + 1e-5f);
}

// ---- residual combine with BN on either operand -------------------------
__global__ void gnn_combine(const float* __restrict__ A, const float* __restrict__ Bm,
                            const float* __restrict__ mean, const float* __restrict__ scale,
                            const float* __restrict__ beta, float* __restrict__ OUT,
                            long long NE, int bn_on_a, int do_relu)
{
  long long i = (long long)blockIdx.x * blockDim.x + threadIdx.x;
  if (i >= NE) return;
  int c = (int)(i & 63);
  float a = A[i], b = Bm[i], r;
  if (bn_on_a) r = (a - mean[c]) * scale[c] + beta[c] + b;
  else         r = a + (b - mean[c]) * scale[c] + beta[c];
  if (do_relu) r = fmaxf(r, 0.f);
  OUT[i] = r;
}

// ---- SAGPool message: MSG[d] += X[s] over valid edges (wave per edge) ---
__global__ __launch_bounds__(256) void gnn_pool_msg(
    const int* __restrict__ src, const int* __restrict__ dst,
    const unsigned char* __restrict__ mask,
    const float* __restrict__ X, float* __restrict__ MSG, int E)
{
  int lane = threadIdx.x & 31;
  int i = (int)((blockIdx.x * (long long)blockDim.x + threadIdx.x) >> 5);
  if (i >= E) return;
  if (mask && !mask[i]) return;
  int s = src[i], d = dst[i];
  const float* xs = X + (size_t)s * HIDDIM;
  float* md = MSG + (size_t)d * HIDDIM;
  atomicAdd(&md[lane],      xs[lane]);
  atomicAdd(&md[lane + 32], xs[lane + 32]);
}

// ---- score = MSG@Wr + br + X@Wo (wave per node) -------------------------
__global__ __launch_bounds__(256) void gnn_score(
    const float* __restrict__ MSG, const float* __restrict__ Wr, const float* __restrict__ br,
    const float* __restrict__ X, const float* __restrict__ Wo,
    float* __restrict__ score, int N)
{
  int lane = threadIdx.x & 31;
  int node = (int)((blockIdx.x * (long long)blockDim.x + threadIdx.x) >> 5);
  if (node >= N) return;
  const float* mr = MSG + (size_t)node * HIDDIM;
  const float* xr = X + (size_t)node * HIDDIM;
  float s = mr[lane] * Wr[lane] + mr[lane + 32] * Wr[lane + 32]
          + xr[lane] * Wo[lane] + xr[lane + 32] * Wo[lane + 32];
#pragma unroll
  for (int off = 16; off > 0; off >>= 1) s += __shfl_xor(s, off, 32);
  if (lane == 0) score[node] = s + br[0];
}

// ---- per-graph bitonic top-k (block per graph, LDS sort) ----------------
extern __shared__ unsigned char gnn_smem[];
__global__ void gnn_topk_sort(const float* __restrict__ score, int n_per, int k,
                              int* __restrict__ perm, int* __restrict__ new_idx)
{
  float* skey = (float*)gnn_smem;
  int*   sidx = (int*)(gnn_smem + (size_t)n_per * sizeof(float));
  int g = blockIdx.x;
  int T = blockDim.x;    // n_per/2
  const float* sc = score + (size_t)g * n_per;
  for (int i = threadIdx.x; i < n_per; i += T) { skey[i] = sc[i]; sidx[i] = i; }
  __syncthreads();
  for (int kk = 2; kk <= n_per; kk <<= 1) {
    for (int j = kk >> 1; j > 0; j >>= 1) {
      for (int t = threadIdx.x; t < n_per; t += T) {
        int ixj = t ^ j;
        if (ixj > t) {
          bool up = ((t & kk) == 0);
          float ka = skey[t], kb = skey[ixj];
          int   ia = sidx[t], ib = sidx[ixj];
          // descending by key, ties -> lower index first (matches lax.top_k)
          bool aFirst = (ka > kb) || (ka == kb && ia < ib);
          if (up ? !aFirst : aFirst) {
            skey[t] = kb; skey[ixj] = ka;
            sidx[t] = ib; sidx[ixj] = ia;
          }
        }
      }
      __syncthreads();
    }
  }
  for (int j = threadIdx.x; j < k; j += T) {
    int node = g * n_per + sidx[j];
    perm[g * k + j] = node;
    new_idx[node] = g * k + j;
  }
}

// ---- gather: XN[i] = X[perm[i]] * tanh(score[perm[i]]) ------------------
__global__ __launch_bounds__(256) void gnn_gather(
    const int* __restrict__ perm, const float* __restrict__ score,
    const float* __restrict__ X, float* __restrict__ XN, int M)
{
  int lane = threadIdx.x & 31;
  int i = (int)((blockIdx.x * (long long)blockDim.x + threadIdx.x) >> 5);
  if (i >= M) return;
  int p = perm[i];
  float t = tanhf(score[p]);
  const float* xs = X + (size_t)p * HIDDIM;
  float* xd = XN + (size_t)i * HIDDIM;
  xd[lane]      = xs[lane] * t;
  xd[lane + 32] = xs[lane + 32] * t;
}

// ---- edge relabel (in-place safe: each thread touches only index e) -----
__global__ void gnn_relabel(const int* __restrict__ srcIn, const int* __restrict__ dstIn,
                            const unsigned char* __restrict__ maskIn,
                            const int* __restrict__ new_idx,
                            int* __restrict__ srcOut, int* __restrict__ dstOut,
                            unsigned char* __restrict__ maskOut, int E)
{
  int e = blockIdx.x * blockDim.x + threadIdx.x;
  if (e >= E) return;
  int s = srcIn[e], d = dstIn[e];
  bool m = maskIn ? (maskIn[e] != 0) : true;
  int s2 = new_idx[s], d2 = new_idx[d];
  bool valid = m && (s2 >= 0) && (d2 >= 0);
  srcOut[e] = valid ? s2 : 0;
  dstOut[e] = valid ? d2 : 0;
  maskOut[e] = valid ? 1 : 0;
}

// ---- final mean pool: out[g,c] = mean over n_per nodes ------------------
__global__ void gnn_mean_pool(const float* __restrict__ X, float* __restrict__ out, int n_per)
{
  int t = blockIdx.x * blockDim.x + threadIdx.x;
  if (t >= NGRAPH * HIDDIM) return;
  int g = t >> 6, c = t & 63;
  const float* base = X + (size_t)g * n_per * HIDDIM + c;
  float s = 0.f;
  for (int r = 0; r < n_per; ++r) s += base[(size_t)r * HIDDIM];
  out[t] = s / (float)n_per;
}

// ===========================================================================
extern "C" void kernel_launch(void* const* d_in, const int* in_sizes, int n_in,
                              void* d_out, int out_size, void* d_ws, size_t ws_size,
                              hipStream_t stream)
{
  (void)n_in; (void)out_size; (void)ws_size;
  constexpr int N0 = 65536;          // 32 graphs * 2048 nodes
  constexpr int EE = N0 * 16;        // 1,048,576 edges

  const float* x_feat = (const float*)d_in[0];
  const int*   ei     = (const int*)d_in[1];   // [2, E]
  const int*   e_src  = ei;
  const int*   e_dst  = ei + EE;

  // ---- parameter mapping (handles both insertion-order and sorted order) --
  auto in = [&](int i) { return (const float*)d_in[4 + i]; };
  const float *W1, *b1, *W2, *b2, *W3, *b3;
  const float *Wg[3], *as_[3], *ad_[3], *bg_[3], *g_[3], *be_[3], *Wr_[3], *br_[3], *Wo_[3];
  if (in_sizes[5] == 4096) {  // tree_flatten sorted-key order: idx5 = W2
    W1 = in(0); W2 = in(1); W3 = in(2);
    Wg[0] = in(3);  Wg[1] = in(4);  Wg[2] = in(5);
    Wo_[0] = in(6); Wo_[1] = in(7); Wo_[2] = in(8);
    Wr_[0] = in(9); Wr_[1] = in(10); Wr_[2] = in(11);
    ad_[0] = in(12); ad_[1] = in(13); ad_[2] = in(14);
    as_[0] = in(15); as_[1] = in(16); as_[2] = in(17);
    b1 = in(18); b2 = in(19); b3 = in(20);
    be_[0] = in(21); be_[1] = in(22); be_[2] = in(23);
    bg_[0] = in(24); bg_[1] = in(25); bg_[2] = in(26);
    br_[0] = in(27); br_[1] = in(28); br_[2] = in(29);
    g_[0] = in(30); g_[1] = in(31); g_[2] = in(32);
  } else {                    // insertion order: idx5 = b1 (64)
    W1 = in(0); b1 = in(1); W2 = in(2); b2 = in(3); W3 = in(4); b3 = in(5);
    for (int i = 0; i < 3; ++i) {
      int o = 6 + i * 9;
      Wg[i] = in(o); as_[i] = in(o + 1); ad_[i] = in(o + 2); bg_[i] = in(o + 3);
      g_[i] = in(o + 4); be_[i] = in(o + 5); Wr_[i] = in(o + 6); br_[i] = in(o + 7); Wo_[i] = in(o + 8);
    }
  }
  const float* Wlin[3] = {W1, W2, W3};
  const float* blin[3] = {b1, b2, b3};

  // ---- workspace carve (~75 MB) -------------------------------------------
  char* w = (char*)d_ws;
  size_t off = 0;
  auto carve = [&](size_t bytes) -> void* {
    void* p = w + off;
    off += (bytes + 255) & ~(size_t)255;
    return p;
  };
  const size_t BIG = (size_t)N0 * HIDDIM * sizeof(float);   // 16 MB
  float* bufA = (float*)carve(BIG);
  float* bufB = (float*)carve(BIG);
  float* bufC = (float*)carve(BIG);
  float* bufD = (float*)carve(BIG);
  float* es_v = (float*)carve((size_t)N0 * 4);
  float* ed_v = (float*)carve((size_t)N0 * 4);
  float* denom = (float*)carve((size_t)N0 * 4);
  float* score = (float*)carve((size_t)N0 * 4);
  unsigned* emax = (unsigned*)carve((size_t)N0 * 4);
  float* pS = (float*)carve((size_t)(N0 / 256) * 64 * 4);
  float* pQ = (float*)carve((size_t)(N0 / 256) * 64 * 4);
  float* bn_mean = (float*)carve(64 * 4);
  float* bn_scale = (float*)carve(64 * 4);
  int* new_idx = (int*)carve((size_t)N0 * 4);
  int* perm = (int*)carve((size_t)(N0 / 2) * 4);
  int* srcW = (int*)carve((size_t)EE * 4);
  int* dstW = (int*)carve((size_t)EE * 4);
  unsigned char* maskW = (unsigned char*)carve((size_t)EE);

  auto cdiv = [](long long a, long long b) { return (unsigned)((a + b - 1) / b); };

  const int nper_in[3] = {2048, 1024, 512};
  float* xlin_b[3]  = {bufB, bufA, bufB};
  float* xpost_b[3] = {bufA, bufB, bufA};

  for (int blk = 0; blk < 3; ++blk) {
    const int nper = nper_in[blk];
    const int N    = NGRAPH * nper;          // nodes this block
    const int K    = (blk == 0) ? 512 : HIDDIM;
    const long long NE = (long long)N * HIDDIM;
    const float* xin = (blk == 0) ? x_feat : (const float*)xpost_b[(blk - 1 + 3) % 3]; // set below
    // actual input: block0 = x_feat, else previous block's gather output
    if (blk == 1) xin = bufB;  // gather(blk0) wrote bufB
    if (blk == 2) xin = bufA;  // gather(blk1) wrote bufA
    float* xlin  = xlin_b[blk];
    float* xl    = bufC;
    float* xg    = bufD;
    float* xpost = xpost_b[blk];
    const int* srcP = (blk == 0) ? e_src : srcW;
    const int* dstP = (blk == 0) ? e_dst : dstW;
    const unsigned char* maskP = (blk == 0) ? nullptr : maskW;
    const int Etot = EE + N;   // real edges + self-loops

    // x = xin @ W + b
    gnn_gemm_n64<<<dim3(cdiv(N, 128)), 256, 0, stream>>>(xin, Wlin[blk], blin[blk], xlin, N, K);
    // xl = x @ Wg (no bias)
    gnn_gemm_n64<<<dim3(cdiv(N, 128)), 256, 0, stream>>>(xlin, Wg[blk], nullptr, xl, N, HIDDIM);
    // per-node attention logits
    gnn_dual_dot<<<dim3(cdiv((long long)N * 32, 256)), 256, 0, stream>>>(xl, as_[blk], ad_[blk], es_v, ed_v, N);
    // init segment buffers
    gnn_fill_u32<<<dim3(cdiv(N, 256)), 256, 0, stream>>>(emax, 0u, N);
    gnn_fill_f32<<<dim3(cdiv(N, 256)), 256, 0, stream>>>(denom, 0.f, N);
    gnn_fill_f32<<<dim3(cdiv(NE, 256)), 256, 0, stream>>>(xg, 0.f, NE);
    // edge softmax
    gnn_edge_max<<<dim3(cdiv(Etot, 256)), 256, 0, stream>>>(srcP, dstP, maskP, es_v, ed_v, emax, EE, N);
    gnn_edge_accum<<<dim3(cdiv(Etot, 8)), 256, 0, stream>>>(srcP, dstP, maskP, es_v, ed_v, emax, xl, denom, xg, EE, N);
    gnn_gat_finalize<<<dim3(cdiv(NE, 256)), 256, 0, stream>>>(xg, denom, bg_[blk], NE);
    // BatchNorm target: block0 -> xg, blocks 1/2 -> xlin
    const float* bn_src = (blk == 0) ? xg : xlin;
    int nb = N / 256;
    gnn_bn_partial<<<dim3((unsigned)nb), 256, 0, stream>>>(bn_src, pS, pQ, N);
    gnn_bn_finalize<<<dim3(1), 64, 0, stream>>>(pS, pQ, nb, N, g_[blk], bn_mean, bn_scale);
    // residual combine (+ReLU except block 3)
    gnn_combine<<<dim3(cdiv(NE, 256)), 256, 0, stream>>>(
        xlin, xg, bn_mean, bn_scale, be_[blk], xpost, NE,
        /*bn_on_a=*/(blk == 0) ? 0 : 1, /*relu=*/(blk == 2) ? 0 : 1);

    // ---- SAGPool ----
    float* msg = bufC;  // xl dead
    gnn_fill_f32<<<dim3(cdiv(NE, 256)), 256, 0, stream>>>(msg, 0.f, NE);
    gnn_pool_msg<<<dim3(cdiv(EE, 8)), 256, 0, stream>>>(srcP, dstP, maskP, xpost, msg, EE);
    gnn_score<<<dim3(cdiv((long long)N * 32, 256)), 256, 0, stream>>>(msg, Wr_[blk], br_[blk], xpost, Wo_[blk], score, N);
    gnn_fill_i32<<<dim3(cdiv(N, 256)), 256, 0, stream>>>(new_idx, -1, N);
    const int k = nper / 2;
    gnn_topk_sort<<<dim3(NGRAPH), nper / 2, (size_t)nper * 8, stream>>>(score, nper, k, perm, new_idx);
    float* xnew = xlin;  // xlin dead after combine
    const int Nnew = NGRAPH * k;
    gnn_gather<<<dim3(cdiv((long long)Nnew * 32, 256)), 256, 0, stream>>>(perm, score, xpost, xnew, Nnew);
    if (blk < 2) {  // block 3's relabeled edges are never used
      gnn_relabel<<<dim3(cdiv(EE, 256)), 256, 0, stream>>>(srcP, dstP, maskP, new_idx, srcW, dstW, maskW, EE);
    }
  }

  // final x lives in gather output of block 3 = bufB ; n_per = 256
  gnn_mean_pool<<<dim3(cdiv(NGRAPH * HIDDIM, 256)), 256, 0, stream>>>(bufB, (float*)d_out, 256);
}